// cnn3dsnn_grating_mem_35734127903358
// MI455X (gfx1250) — compile-verified
//
#include <hip/hip_runtime.h>
#include <stdint.h>

typedef __attribute__((ext_vector_type(2))) float v2f;
typedef __attribute__((ext_vector_type(8))) float v8f;

// D = A(16x4) * B(4x16) + C, fp32, one wave.
__device__ __forceinline__ v8f wmma4(v2f a, v2f b, v8f c) {
  return __builtin_amdgcn_wmma_f32_16x16x4_f32(false, a, false, b, (short)0, c,
                                               false, false);
}

// ---------------------------------------------------------------- pool ----
__global__ void pool2x2_kernel(const float* __restrict__ x,
                               float* __restrict__ xp, int total) {
  int i = blockIdx.x * blockDim.x + threadIdx.x;
  if (i >= total) return;
  int c = i % 301;
  int t = i / 301;
  int r = t % 301;
  int bc = t / 301;
  const float* src = x + ((size_t)bc * 602 + 2 * r) * 602 + 2 * c;
  xp[i] = 0.25f * (src[0] + src[1] + src[602] + src[603]);
}

// -------------------------------------------- weight panel preparation ----
// bpad[((ci*3+kd)*15+kh)*16 + kw][co], kw padded 15->16 with zeros: no K
// guards in the hot loop and B loads are lane-coalesced.
__global__ void conv_wprep_kernel(const float* __restrict__ wgt,
                                  float* __restrict__ bpad, int Cout, int Cin,
                                  int total) {
  int i = blockIdx.x * blockDim.x + threadIdx.x;
  if (i >= total) return;
  int co = i % Cout;
  int k = i / Cout;
  int kw = k & 15;
  int row = k >> 4;  // (ci*3+kd)*15+kh
  float v = 0.f;
  if (kw < 15) v = wgt[(size_t)co * (Cin * 675) + row * 15 + kw];
  bpad[i] = v;
}

// ------------------------------------------------- conv3d as WMMA GEMM ----
// One wave: 16 out-positions (W) x 16 out-channels. Uniform SGPR bases +
// one 32-bit lane offset per stream -> saddr-form loads, no 64-bit VALU.
template <int CIN, int DIN, int HIN, int WIN, int COUT, int DOUT, int HOUT,
          int WOUT>
__global__ __launch_bounds__(32) void conv3d_wmma_kernel(
    const float* __restrict__ in, const float* __restrict__ bpad,
    const float* __restrict__ bias, float* __restrict__ out) {
  constexpr int HW = HIN * WIN;
  constexpr int WT = (WOUT + 15) / 16;
  constexpr int CT = COUT / 16;

  int bid = blockIdx.x;
  int cot = bid % CT;   bid /= CT;
  int wt  = bid % WT;   bid /= WT;
  int ho  = bid % HOUT; bid /= HOUT;
  int dho = bid % DOUT; bid /= DOUT;
  int bb  = bid;

  int lane = threadIdx.x;
  int mm = lane & 15;  // A row / B col owned by this lane
  int lh = lane >> 4;  // lane half -> K {0,1} vs {2,3} of each chunk

  int w0 = wt << 4;
  int wA = w0 + mm;
  if (wA >= WOUT) wA = WOUT - 1;  // clamped duplicate rows, stores masked
  int nCo = (cot << 4) + mm;

  int aoff  = 2 * wA;            // per-lane element offset within an A row
  int aoffl = aoff + 2 * lh;     // + lane-half K offset
  int boff  = nCo + 2 * lh * COUT;  // per-lane offset within a B k-row pair

  const float* abase =
      in + (size_t)bb * (CIN * DIN * HW) + (size_t)(2 * ho) * WIN;  // uniform
  const float* brow = bpad;                                         // uniform

  v8f acc = {};
  for (int ci = 0; ci < CIN; ++ci) {
    for (int kd = 0; kd < 3; ++kd) {
      const float* arow = abase + (ci * DIN + 2 * dho + kd) * HW;  // uniform
#pragma unroll 2
      for (int kh = 0; kh < 15; ++kh) {
#pragma unroll
        for (int c = 0; c < 4; ++c) {
          v2f av;
          av.x = arow[aoffl + 4 * c];
          if (c < 3) {
            av.y = arow[aoffl + 4 * c + 1];
          } else {  // kw==15 is the zero pad slot; keep load in-bounds
            int idx = (lh == 0) ? (aoff + 14) : aoff;
            float t = arow[idx];
            av.y = (lh == 0) ? t : 0.f;
          }
          v2f bv;
          bv.x = brow[boff + (4 * c) * COUT];      // coalesced across lanes
          bv.y = brow[boff + (4 * c + 1) * COUT];  // zero column at kw==15
          acc = wmma4(av, bv, acc);
        }
        arow += WIN;
        brow += 16 * COUT;
      }
    }
  }

  float bsc = bias[nCo];
  size_t outbase = (((size_t)bb * COUT + nCo) * DOUT + dho) *
                       (size_t)(HOUT * WOUT) +
                   (size_t)ho * WOUT;
  for (int r = 0; r < 8; ++r) {
    int M = r + (lh << 3);  // C/D layout: lane half = M+8
    int wo = w0 + M;
    if (wo < WOUT) out[outbase + wo] = fmaxf(acc[r] + bsc, 0.0f);
  }
}

// -------------------------------------------------- Threefry-2x32 (JAX) ---
__device__ __forceinline__ uint32_t rotl32(uint32_t x, int n) {
  return (x << n) | (x >> (32 - n));
}
__device__ __forceinline__ void threefry2x32(uint32_t k0, uint32_t k1,
                                             uint32_t x0, uint32_t x1,
                                             uint32_t& o0, uint32_t& o1) {
  uint32_t k2 = k0 ^ k1 ^ 0x1BD11BDAu;
  x0 += k0; x1 += k1;
#define TFR(r) { x0 += x1; x1 = rotl32(x1, (r)); x1 ^= x0; }
  TFR(13) TFR(15) TFR(26) TFR(6)  x0 += k1; x1 += k2 + 1u;
  TFR(17) TFR(29) TFR(16) TFR(24) x0 += k2; x1 += k0 + 2u;
  TFR(13) TFR(15) TFR(26) TFR(6)  x0 += k0; x1 += k1 + 3u;
  TFR(17) TFR(29) TFR(16) TFR(24) x0 += k1; x1 += k2 + 4u;
  TFR(13) TFR(15) TFR(26) TFR(6)  x0 += k2; x1 += k0 + 5u;
#undef TFR
  o0 = x0; o1 = x1;
}

// jax.random.split(key(42), 30): bits = threefry(key, iota(60)) with halved
// counters; keys[t] = (bits[2t], bits[2t+1]).
__global__ void snn_keys_kernel(uint32_t* __restrict__ keys) {
  int t = threadIdx.x;
  if (t >= 30) return;
  for (int j = 0; j < 2; ++j) {
    uint32_t idx = 2u * t + j;
    uint32_t a, b;
    if (idx < 30u) { threefry2x32(0u, 42u, idx, idx + 30u, a, b); keys[2 * t + j] = a; }
    else           { threefry2x32(0u, 42u, idx - 30u, idx, a, b); keys[2 * t + j] = b; }
  }
}

// spikes[t][b][k] in {-1,0,1} as float; exact JAX uniform replication.
__global__ void spike_kernel(const float* __restrict__ xsnn,
                             const uint32_t* __restrict__ keys,
                             float* __restrict__ spikes, int total) {
  int i = blockIdx.x * blockDim.x + threadIdx.x;
  if (i >= total) return;
  const int NE = 86528, HALFN = 43264;  // 4*21632 per step
  int t = i / NE;
  int e = i - t * NE;
  uint32_t k0 = keys[2 * t], k1 = keys[2 * t + 1];
  uint32_t a, b, bits;
  if (e < HALFN) { threefry2x32(k0, k1, (uint32_t)e, (uint32_t)(e + HALFN), a, b); bits = a; }
  else           { threefry2x32(k0, k1, (uint32_t)(e - HALFN), (uint32_t)e, a, b); bits = b; }
  float r = __uint_as_float((bits >> 9) | 0x3f800000u) - 1.0f;
  float xv = xsnn[e];
  float s = 0.0f;
  if (0.5f * fabsf(xv) > r) s = (xv > 0.f) ? 1.0f : ((xv < 0.f) ? -1.0f : 0.0f);
  spikes[i] = s;
}

// ---------------- batched fc1: C[m=(t*4+b)][n] = spikes@W1.T + b1 ----------
__global__ __launch_bounds__(32) void fc1_gemm_kernel(
    const float* __restrict__ spikes,  // (120 rows x 21632)
    const float* __restrict__ w1,      // (512, 21632)
    const float* __restrict__ b1,      // (512)
    float* __restrict__ C) {           // (128, 512); rows <120 written
  const int K = 21632, N = 512, MV = 120;
  int mt = blockIdx.x >> 5;  // 8 M-tiles
  int nt = blockIdx.x & 31;  // 32 N-tiles
  int lane = threadIdx.x;
  int mm = lane & 15;
  int lh = lane >> 4;
  int mload = (mt << 4) + mm;
  if (mload >= MV) mload = MV - 1;  // duplicate row: affects only D rows >=120
  int nload = (nt << 4) + mm;

  // 32-bit v2f indices off uniform bases -> saddr-form loads.
  const v2f* sp2 = reinterpret_cast<const v2f*>(spikes);
  const v2f* wp2 = reinterpret_cast<const v2f*>(w1);
  int aoff2 = mload * (K / 2) + lh;
  int boff2 = nload * (K / 2) + lh;

  v8f acc = {};
  for (int kg2 = 0; kg2 < K / 2; kg2 += 8) {  // 8 v2f = 16 floats per group
    __builtin_prefetch(wp2 + boff2 + kg2 + 256, 0, 1);  // global_prefetch_b8
#pragma unroll
    for (int u = 0; u < 4; ++u) {
      v2f av = sp2[aoff2 + kg2 + 2 * u];
      v2f bv = wp2[boff2 + kg2 + 2 * u];
      acc = wmma4(av, bv, acc);
    }
  }
  float bias = b1[nload];
  for (int r = 0; r < 8; ++r) {
    int M = (mt << 4) + r + (lh << 3);
    if (M < MV) C[M * N + nload] = acc[r] + bias;
  }
}

// --------------- sequential SNN dynamics (threshold/reset + fc2) ----------
__global__ __launch_bounds__(512) void snn_seq_kernel(
    const float* __restrict__ C,   // (>=120, 512), row = t*4+b
    const float* __restrict__ w2,  // (54, 512)
    const float* __restrict__ b2,  // (54)
    float* __restrict__ out) {     // (4, 54)
  __shared__ float sh_out[4][512];
  int tid = threadIdx.x;
  float m0[4] = {0.f, 0.f, 0.f, 0.f};
  float m1 = 0.0f;
  int bj_b = tid / 54, bj_j = tid % 54;  // valid when tid < 216
  for (int t = 0; t < 30; ++t) {
    for (int b = 0; b < 4; ++b) {
      float v = m0[b] + C[(t * 4 + b) * 512 + tid];
      bool fire = v > 1.0f;
      sh_out[b][tid] = fire ? 1.0f : 0.0f;
      m0[b] = fire ? 0.0f : v;
    }
    __syncthreads();
    if (tid < 216) {
      float acc = b2[bj_j];
      const float* wr = w2 + bj_j * 512;
      for (int n = 0; n < 512; ++n) acc += sh_out[bj_b][n] * wr[n];
      m1 += acc;
    }
    __syncthreads();
  }
  if (tid < 216) out[bj_b * 54 + bj_j] = m1;
}

// --------------------------------------------------------------- launch ---
extern "C" void kernel_launch(void* const* d_in, const int* in_sizes, int n_in,
                              void* d_out, int out_size, void* d_ws,
                              size_t ws_size, hipStream_t stream) {
  const float* x    = (const float*)d_in[0];
  const float* w1   = (const float*)d_in[1];
  const float* b1   = (const float*)d_in[2];
  const float* w2   = (const float*)d_in[3];
  const float* b2   = (const float*)d_in[4];
  const float* w3   = (const float*)d_in[5];
  const float* b3   = (const float*)d_in[6];
  const float* fc1w = (const float*)d_in[7];
  const float* fc1b = (const float*)d_in[8];
  const float* fc2w = (const float*)d_in[9];
  const float* fc2b = (const float*)d_in[10];
  float* out = (float*)d_out;

  float* ws = (float*)d_ws;
  size_t off = 0;
  float* xp = ws + off;    off += (size_t)4 * 24 * 301 * 301;
  float* h1 = ws + off;    off += (size_t)4 * 16 * 11 * 144 * 144;
  float* h2 = ws + off;    off += (size_t)4 * 32 * 5 * 65 * 65;
  float* h3 = ws + off;    off += (size_t)4 * 21632;  // x_snn
  float* spk = ws + off;   off += (size_t)30 * 4 * 21632;
  float* Cbuf = ws + off;  off += (size_t)128 * 512;
  float* bpad1 = ws + off; off += (size_t)1 * 720 * 16;
  float* bpad2 = ws + off; off += (size_t)16 * 720 * 32;
  float* bpad3 = ws + off; off += (size_t)32 * 720 * 16;
  uint32_t* keys = (uint32_t*)(ws + off);  // 60 uints

  {  // pool 2x2
    int total = 4 * 24 * 301 * 301;
    pool2x2_kernel<<<(total + 255) / 256, 256, 0, stream>>>(x, xp, total);
  }
  {  // padded/transposed weight panels
    int t1 = 1 * 720 * 16;
    conv_wprep_kernel<<<(t1 + 255) / 256, 256, 0, stream>>>(w1, bpad1, 16, 1, t1);
    int t2 = 16 * 720 * 32;
    conv_wprep_kernel<<<(t2 + 255) / 256, 256, 0, stream>>>(w2, bpad2, 32, 16, t2);
    int t3 = 32 * 720 * 16;
    conv_wprep_kernel<<<(t3 + 255) / 256, 256, 0, stream>>>(w3, bpad3, 16, 32, t3);
  }
  // conv1: (4,1,24,301,301) -> (4,16,11,144,144); wtiles=9, cotiles=1
  conv3d_wmma_kernel<1, 24, 301, 301, 16, 11, 144, 144>
      <<<4 * 11 * 144 * 9 * 1, 32, 0, stream>>>(xp, bpad1, b1, h1);
  // conv2: -> (4,32,5,65,65); wtiles=5, cotiles=2
  conv3d_wmma_kernel<16, 11, 144, 144, 32, 5, 65, 65>
      <<<4 * 5 * 65 * 5 * 2, 32, 0, stream>>>(h1, bpad2, b2, h2);
  // conv3: -> (4,16,2,26,26); wtiles=2, cotiles=1
  conv3d_wmma_kernel<32, 5, 65, 65, 16, 2, 26, 26>
      <<<4 * 2 * 26 * 2 * 1, 32, 0, stream>>>(h2, bpad3, b3, h3);

  snn_keys_kernel<<<1, 32, 0, stream>>>(keys);
  {
    int total = 30 * 4 * 21632;
    spike_kernel<<<(total + 255) / 256, 256, 0, stream>>>(h3, keys, spk, total);
  }
  fc1_gemm_kernel<<<256, 32, 0, stream>>>(spk, fc1w, fc1b, Cbuf);
  snn_seq_kernel<<<1, 512, 0, stream>>>(Cbuf, fc2w, fc2b, out);
}